// VQEModel_67912022884828
// MI455X (gfx1250) — compile-verified
//
#include <hip/hip_runtime.h>
#include <hip/hip_bf16.h>

typedef float v2f __attribute__((ext_vector_type(2)));
typedef float v8f __attribute__((ext_vector_type(8)));

#define NQ     13
#define DIM    8192          // 2^13
#define LAYERS 2

// ---------------------------------------------------------------------------
// Kernel 1: ansatz. One workgroup, whole state in LDS (64 KB of 320 KB/WGP).
// Applies RX sweeps, collapsed RZ (13 identical diagonal gates -> one pass),
// CRX+SWAP pyramid (merged into one pass per gate pair), SWAP pyramid.
// Writes psi (real/imag) to workspace and the Z-marginal term to partials[512].
// ---------------------------------------------------------------------------
__global__ __launch_bounds__(512)
void vqe_ansatz(const float* __restrict__ theta, const float* __restrict__ phi,
                const float* __restrict__ alpha,
                float* __restrict__ wa, float* __restrict__ wb,
                float* __restrict__ partials) {
  __shared__ float sr[DIM];
  __shared__ float si[DIM];
  __shared__ float red[512];
  const int tid = threadIdx.x;

  for (int s = tid; s < DIM; s += 512) { sr[s] = (s == 0) ? 1.0f : 0.0f; si[s] = 0.0f; }
  __syncthreads();

  for (int layer = 0; layer < LAYERS; ++layer) {
    const float th = theta[layer];
    const float ph = phi[layer];
    const float al = alpha[layer];
    const float c1 = cosf(0.5f * th), s1 = sinf(0.5f * th);

    // RX on every qubit (axis q -> bit p = 12-q; axis 0 is MSB of flat index)
    for (int q = 0; q < NQ; ++q) {
      const int p = NQ - 1 - q;
      const int mask = (1 << p) - 1;
      for (int idx = tid; idx < DIM / 2; idx += 512) {
        const int low = idx & mask;
        const int s0  = ((idx >> p) << (p + 1)) | low;
        const int s1i = s0 | (1 << p);
        const float a0r = sr[s0],  a0i = si[s0];
        const float a1r = sr[s1i], a1i = si[s1i];
        // new0 = c*a0 - i*s*a1 ; new1 = -i*s*a0 + c*a1
        sr[s0]  = c1 * a0r + s1 * a1i;   si[s0]  = c1 * a0i - s1 * a1r;
        sr[s1i] = s1 * a0i + c1 * a1r;   si[s1i] = -s1 * a0r + c1 * a1i;
      }
      __syncthreads();
    }

    // RZ on every qubit, collapsed: phase = (phi/2)*(2*popcount(s) - 13)
    for (int s = tid; s < DIM; s += 512) {
      const float ang = 0.5f * ph * (float)(2 * __popc(s) - NQ);
      const float cz = cosf(ang), sz = sinf(ang);
      const float r = sr[s], m = si[s];
      sr[s] = r * cz - m * sz;
      si[s] = r * sz + m * cz;
    }
    __syncthreads();

    const float ca = cosf(0.5f * al), sa = sinf(0.5f * al);

    // CRX(k,k+1) followed by SWAP(k,k+1), merged into one LDS pass per pair.
    // control axis k -> bit pT+1, target axis k+1 -> bit pT = 11-k.
    for (int j = 0; j < NQ - 1; ++j) {
      for (int k = 0; k < NQ - 1 - j; ++k) {
        const int pT = NQ - 2 - k;
        const int maskT = (1 << pT) - 1;
        for (int g = tid; g < DIM / 4; g += 512) {
          const int low  = g & maskT;
          const int base = ((g >> pT) << (pT + 2)) | low;
          const int i01 = base | (1 << pT);
          const int i10 = base | (2 << pT);
          const int i11 = base | (3 << pT);
          const float r10 = sr[i10], m10 = si[i10];
          const float r11 = sr[i11], m11 = si[i11];
          // CRX on control=1 block: n10 = c*a10 - i*s*a11 ; n11 = -i*s*a10 + c*a11
          const float n10r = ca * r10 + sa * m11, n10i = ca * m10 - sa * r11;
          const float n11r = sa * m10 + ca * r11, n11i = -sa * r10 + ca * m11;
          const float r01 = sr[i01], m01 = si[i01];
          // then SWAP: a01' = n10 ; a10' = old a01 ; a11' = n11
          sr[i01] = n10r; si[i01] = n10i;
          sr[i10] = r01;  si[i10] = m01;
          sr[i11] = n11r; si[i11] = n11i;
        }
        __syncthreads();
      }
    }

    // pure SWAP pyramid
    for (int j = 0; j < NQ - 1; ++j) {
      for (int k = 0; k < NQ - 1 - j; ++k) {
        const int pT = NQ - 2 - k;
        const int maskT = (1 << pT) - 1;
        for (int g = tid; g < DIM / 4; g += 512) {
          const int low  = g & maskT;
          const int base = ((g >> pT) << (pT + 2)) | low;
          const int i01 = base | (1 << pT);
          const int i10 = base | (2 << pT);
          const float r01 = sr[i01], m01 = si[i01];
          sr[i01] = sr[i10]; si[i01] = si[i10];
          sr[i10] = r01;     si[i10] = m01;
        }
        __syncthreads();
      }
    }
  }

  // write psi to workspace; fold all 13 Z-expectations into one weighted sum
  float z = 0.0f;
  for (int s = tid; s < DIM; s += 512) {
    const float a = sr[s], b = si[s];
    wa[s] = a; wb[s] = b;
    z += (a * a + b * b) * (float)(NQ - 2 * __popc(s));
  }
  red[tid] = z;
  __syncthreads();
  for (int off = 256; off > 0; off >>= 1) {
    if (tid < off) red[tid] += red[tid + off];
    __syncthreads();
  }
  if (tid == 0) partials[512] = -0.1f * red[0];
}

// ---------------------------------------------------------------------------
// Kernel 2: energy contraction Re(psi^H H psi), HBM-bound (512 MB of H read
// exactly once). Each block owns a 16-row block; its 8 waves own 1024-wide K
// chunks. Per K-step of 4: two global b64 loads (Hr, Hi tile slices as the
// WMMA A operand), two LDS b64 broadcasts (psi as B operand, all 16 columns
// identical), four V_WMMA_F32_16X16X4_F32 for complex accumulation.
//   yr += Hr*a - Hi*b   (Hi term via negated B: f32 WMMA has no A/B-neg)
//   yi += Hr*b + Hi*a
// A layout (32-bit 16x4): lanes 0-15 -> K=0,1 ; lanes 16-31 -> K=2,3.
// B layout (4x16): VGPR0 -> K=0 | K=2 ; VGPR1 -> K=1 | K=3 (rows striped
// across lanes, second lane-half offset by 2) -> same kb index as A.
// ---------------------------------------------------------------------------
__global__ __launch_bounds__(256)
void vqe_energy(const float* __restrict__ Hr, const float* __restrict__ Hi,
                const float* __restrict__ wa, const float* __restrict__ wb,
                float* __restrict__ partials) {
  __shared__ float sa[DIM];
  __shared__ float sb[DIM];
  __shared__ float red[256];
  const int tid = threadIdx.x;

  for (int s = tid; s < DIM; s += 256) { sa[s] = wa[s]; sb[s] = wb[s]; }
  __syncthreads();

  const int wave     = tid >> 5;        // 0..7 -> K chunk
  const int lane     = tid & 31;
  const int laneHalf = lane >> 4;       // 0: K=0,1 ; 1: K=2,3
  const int laneM    = lane & 15;       // row within 16-row block
  const int row0     = blockIdx.x * 16;
  const int row      = row0 + laneM;
  const int kb       = laneHalf * 2;
  const int j0       = wave * 1024;

  const float* hr = Hr + (size_t)row * DIM + kb;
  const float* hi = Hi + (size_t)row * DIM + kb;

  v8f yr = {};
  v8f yi = {};

  for (int j = j0; j < j0 + 1024; j += 4) {
    const v2f ar = *(const v2f*)(hr + j);                  // Hr[row, j+kb .. +1]
    const v2f ai = *(const v2f*)(hi + j);                  // Hi[row, j+kb .. +1]
    const v2f br = { sa[j + kb], sa[j + kb + 1] };         // psi_re chunk (bcast)
    const v2f bi = { sb[j + kb], sb[j + kb + 1] };         // psi_im chunk (bcast)
    const v2f nbi = -bi;
    yr = __builtin_amdgcn_wmma_f32_16x16x4_f32(false, ar, false, br,  (short)0, yr, false, false);
    yr = __builtin_amdgcn_wmma_f32_16x16x4_f32(false, ai, false, nbi, (short)0, yr, false, false);
    yi = __builtin_amdgcn_wmma_f32_16x16x4_f32(false, ar, false, bi,  (short)0, yi, false, false);
    yi = __builtin_amdgcn_wmma_f32_16x16x4_f32(false, ai, false, br,  (short)0, yi, false, false);
  }

  // D layout: VGPR v -> row v (lanes 0-15) / row v+8 (lanes 16-31); the 16
  // columns are identical, so summing all lanes gives 16x the tile value.
  float contrib = 0.0f;
  const int rbase = row0 + laneHalf * 8;
#pragma unroll
  for (int v = 0; v < 8; ++v)
    contrib += sa[rbase + v] * yr[v] + sb[rbase + v] * yi[v];

  red[tid] = contrib;
  __syncthreads();
  for (int off = 128; off > 0; off >>= 1) {
    if (tid < off) red[tid] += red[tid + off];
    __syncthreads();
  }
  if (tid == 0) partials[blockIdx.x] = red[0] * (1.0f / 16.0f);
}

// ---------------------------------------------------------------------------
// Kernel 3: deterministic fixed-order final reduction of 512 block partials
// + the Z term slot (partials[512]).
// ---------------------------------------------------------------------------
__global__ __launch_bounds__(256)
void vqe_reduce(const float* __restrict__ partials, float* __restrict__ out) {
  __shared__ float red[256];
  const int tid = threadIdx.x;
  float v = partials[tid] + partials[tid + 256];
  if (tid == 0) v += partials[512];
  red[tid] = v;
  __syncthreads();
  for (int off = 128; off > 0; off >>= 1) {
    if (tid < off) red[tid] += red[tid + off];
    __syncthreads();
  }
  if (tid == 0) out[0] = red[0];
}

// ---------------------------------------------------------------------------
extern "C" void kernel_launch(void* const* d_in, const int* in_sizes, int n_in,
                              void* d_out, int out_size, void* d_ws, size_t ws_size,
                              hipStream_t stream) {
  const float* theta = (const float*)d_in[0];
  const float* phi   = (const float*)d_in[1];
  const float* alpha = (const float*)d_in[2];
  const float* Hr    = (const float*)d_in[3];
  const float* Hi    = (const float*)d_in[4];

  float* ws       = (float*)d_ws;
  float* wa       = ws;               // psi real  [8192]
  float* wb       = ws + DIM;         // psi imag  [8192]
  float* partials = ws + 2 * DIM;     // [513]: 512 block partials + Z term

  vqe_ansatz<<<1, 512, 0, stream>>>(theta, phi, alpha, wa, wb, partials);
  vqe_energy<<<512, 256, 0, stream>>>(Hr, Hi, wa, wb, partials);
  vqe_reduce<<<1, 256, 0, stream>>>(partials, (float*)d_out);
}